// CoMA_46248207843712
// MI455X (gfx1250) — compile-verified
//
#include <hip/hip_runtime.h>
#include <hip/hip_bf16.h>

// CDNA5 / gfx1250. wave32. V_WMMA_F32_16X16X4_F32 for all dense GEMM work
// (per-node ChebConv einsum + split-K encoder GEMM), fp32 end to end.
// Memory-bound streaming kernels (scatter/pool/fill/bias) use float4 b128
// accesses; ChebConv weights staged in LDS; all shape constants templated.

typedef __attribute__((ext_vector_type(2))) float v2f;
typedef __attribute__((ext_vector_type(8))) float v8f;

#define WMMA_F32_16x16x4(a, b, c) \
  __builtin_amdgcn_wmma_f32_16x16x4_f32(false, (a), false, (b), (short)0, (c), false, false)

// ---------------------------------------------------------------------------
// (B, N, 3) -> (N, B, 3) node-major transpose of the input point cloud.
__global__ void k_transpose_x(const float* __restrict__ x, float* __restrict__ xt, int N) {
    size_t total = (size_t)N * 32 * 3;
    size_t stride = (size_t)gridDim.x * blockDim.x;
    for (size_t i = (size_t)blockIdx.x * blockDim.x + threadIdx.x; i < total; i += stride) {
        size_t n = i / 96;             // constant divisor -> mulhi
        int r = (int)(i % 96);
        int b = r / 3, f = r % 3;
        xt[i] = x[(size_t)b * N * 3 + n * 3 + f];
    }
}

// n4 = element count / 4 (all buffer sizes are multiples of 4)
__global__ void k_fill4(float4* __restrict__ p, size_t n4, float v) {
    size_t stride = (size_t)gridDim.x * blockDim.x;
    float4 vv = make_float4(v, v, v, v);
    for (size_t i = (size_t)blockIdx.x * blockDim.x + threadIdx.x; i < n4; i += stride)
        p[i] = vv;
}

// dst = s * src  (t2 = -t0 before the scatter adds 2*L*t1)
__global__ void k_scale_copy4(float4* __restrict__ dst, const float4* __restrict__ src,
                              size_t n4, float s) {
    size_t stride = (size_t)gridDim.x * blockDim.x;
    for (size_t i = (size_t)blockIdx.x * blockDim.x + threadIdx.x; i < n4; i += stride) {
        float4 v = src[i];
        dst[i] = make_float4(s * v.x, s * v.y, s * v.z, s * v.w);
    }
}

// Sparse propagate: dst[row[e]] += scale * norm[e] * src[col[e]] over a TILE
// payload per edge. One thread handles 4 consecutive floats: b128 load +
// 4 global_atomic_add_f32 (merged immediate offsets).
template <int TILE>
__global__ void k_scatter(const int* __restrict__ edge, const float* __restrict__ norm,
                          const float* __restrict__ src, float* __restrict__ dst,
                          int E, float scale) {
    constexpr int QT = TILE / 4;
    size_t total = (size_t)E * QT;
    size_t stride = (size_t)gridDim.x * blockDim.x;
    for (size_t i = (size_t)blockIdx.x * blockDim.x + threadIdx.x; i < total; i += stride) {
        int e = (int)(i / QT);
        int q = (int)(i % QT);
        int row = edge[e];
        int col = edge[E + e];
        float s = scale * norm[e];
        float4 v = *(const float4*)(src + (size_t)col * TILE + q * 4);
        float* d = dst + (size_t)row * TILE + q * 4;
        atomicAdd(d + 0, s * v.x);
        atomicAdd(d + 1, s * v.y);
        atomicAdd(d + 2, s * v.z);
        atomicAdd(d + 3, s * v.w);
    }
}

// Per-node GEMM: out[node](32 x Fout) += T[node](32 x Fin) @ Wk(Fin x Fout)
// One wave per (node, batch-half, Fout-16-tile) C tile via fp32 WMMA.
// A 16x4 frag: lane L holds M=L%16, K=(L/16)*2+v. B 4x16: N=L%16, K=(L/16)*2+v.
// C/D 16x16: VGPR r holds M=r+(L/16)*8, N=L%16. Wk staged in LDS.
template <int Fin, int Fout>
__global__ void __launch_bounds__(256) k_cheb_gemm(const float* __restrict__ T,
                                                   const float* __restrict__ Wk,
                                                   float* __restrict__ out, int nNodes) {
    constexpr int NT = Fout / 16;          // 1 or 2 Fout tiles
    constexpr int TPN = 2 * NT;            // tile tasks per node
    __shared__ float sW[Fin * Fout];
    for (int t = threadIdx.x; t < Fin * Fout; t += blockDim.x)
        sW[t] = Wk[t];
    __syncthreads();

    int lane = threadIdx.x & 31;
    int l15 = lane & 15;
    int half = lane >> 4;
    int tasks = nNodes * TPN;
    int wave = (int)(((size_t)blockIdx.x * blockDim.x + threadIdx.x) >> 5);
    int nwaves = (int)(((size_t)gridDim.x * blockDim.x) >> 5);
    for (int task = wave; task < tasks; task += nwaves) {  // wave-uniform trips
        int node = task / TPN;             // constant divisor
        int rem = task - node * TPN;
        int mtile = rem / NT;
        int ntile = rem - mtile * NT;
        const float* Tt = T + (size_t)node * 32 * Fin;
        float* Ot = out + (size_t)node * 32 * Fout;
        int m = mtile * 16 + l15;
        int ncol = ntile * 16 + l15;
        v8f c;
#pragma unroll
        for (int r = 0; r < 8; ++r)
            c[r] = Ot[(size_t)(mtile * 16 + r + half * 8) * Fout + ncol];
#pragma unroll
        for (int kk = 0; kk < Fin; kk += 4) {
            int ka = kk + half * 2;
            v2f a, b;
            if constexpr ((Fin & 3) == 0) {
                a = *(const v2f*)(Tt + (size_t)m * Fin + ka);   // 8B-aligned b64 load
            } else {
                a.x = (ka + 0 < Fin) ? Tt[(size_t)m * Fin + ka + 0] : 0.0f;
                a.y = (ka + 1 < Fin) ? Tt[(size_t)m * Fin + ka + 1] : 0.0f;
            }
            b.x = (ka + 0 < Fin) ? sW[(ka + 0) * Fout + ncol] : 0.0f;
            b.y = (ka + 1 < Fin) ? sW[(ka + 1) * Fout + ncol] : 0.0f;
            c = WMMA_F32_16x16x4(a, b, c);
        }
#pragma unroll
        for (int r = 0; r < 8; ++r)
            Ot[(size_t)(mtile * 16 + r + half * 8) * Fout + ncol] = c[r];
    }
}

// o[i] = relu(o[i] + bias[i % Fout]) vectorized by 4 (Fout is 16/32/128).
template <int Fout>
__global__ void k_bias_relu4(float* __restrict__ o, const float* __restrict__ bias,
                             size_t n4) {
    size_t stride = (size_t)gridDim.x * blockDim.x;
    for (size_t i = (size_t)blockIdx.x * blockDim.x + threadIdx.x; i < n4; i += stride) {
        float4 v = ((const float4*)o)[i];
        float4 bb = *(const float4*)(bias + ((i * 4) & (size_t)(Fout - 1)));
        v.x += bb.x; v.y += bb.y; v.z += bb.z; v.w += bb.w;
        v.x = v.x > 0.0f ? v.x : 0.0f;
        v.y = v.y > 0.0f ? v.y : 0.0f;
        v.z = v.z > 0.0f ? v.z : 0.0f;
        v.w = v.w > 0.0f ? v.w : 0.0f;
        ((float4*)o)[i] = v;
    }
}

// Pool: exactly 3 weighted inputs per output row (pure gather, no atomics).
// pidx is (2, nOut*3): row ids first, col ids at offset nOut*3. float4 payload.
template <int TILE>
__global__ void k_pool(const int* __restrict__ pidx, const float* __restrict__ pval,
                       const float* __restrict__ h, float* __restrict__ nxt, int nOut) {
    constexpr int QT = TILE / 4;
    size_t total = (size_t)nOut * QT;
    size_t stride = (size_t)gridDim.x * blockDim.x;
    const int* cols = pidx + (size_t)nOut * 3;
    for (size_t i = (size_t)blockIdx.x * blockDim.x + threadIdx.x; i < total; i += stride) {
        int r = (int)(i / QT);
        int q = (int)(i % QT);
        float4 acc = make_float4(0.0f, 0.0f, 0.0f, 0.0f);
#pragma unroll
        for (int j = 0; j < 3; ++j) {
            int cidx = cols[r * 3 + j];
            float p = pval[r * 3 + j];
            float4 v = *(const float4*)(h + (size_t)cidx * TILE + q * 4);
            acc.x += p * v.x; acc.y += p * v.y; acc.z += p * v.z; acc.w += p * v.w;
        }
        *(float4*)(nxt + (size_t)r * TILE + q * 4) = acc;
    }
}

// Split-K encoder: acc(32 x 128) += Hflat(32 x 25024) @ Wenc, partial over K.
// Hflat[m][k], k = n*32 + f  ->  H[n*1024 + m*32 + f] (node-major).
// Each block = 16 waves = all (mtile,ntile) C tiles; blocks stride over K and
// atomically accumulate fp32 fragments. Bias+ReLU applied by k_bias_relu4<128>.
__global__ void __launch_bounds__(512) k_enc_partial(const float* __restrict__ H,
                                                     const float* __restrict__ Wenc,
                                                     float* __restrict__ acc) {
    int lane = threadIdx.x & 31;
    int wave = threadIdx.x >> 5;   // 0..15
    int mtile = wave >> 3;         // 0..1
    int ntile = wave & 7;          // 0..7
    int half = lane >> 4;
    int l15 = lane & 15;
    int m = mtile * 16 + l15;
    int ncol = ntile * 16 + l15;
    const int Ktot = 782 * 32;     // 25024, multiple of 4
    int kstride = (int)gridDim.x * 4;
    v8f c = {};
    for (int kk = (int)blockIdx.x * 4; kk < Ktot; kk += kstride) {  // wave-uniform
        int ka = kk + half * 2;    // even -> a-pair contiguous & 8B-aligned
        v2f a, b;
        a = *(const v2f*)(H + (size_t)(ka >> 5) * 1024 + m * 32 + (ka & 31));
        b.x = Wenc[(size_t)(ka + 0) * 128 + ncol];
        b.y = Wenc[(size_t)(ka + 1) * 128 + ncol];
        c = WMMA_F32_16x16x4(a, b, c);
    }
#pragma unroll
    for (int r = 0; r < 8; ++r)
        atomicAdd(&acc[(mtile * 16 + r + half * 8) * 128 + ncol], c[r]);
}

// Classifier: (32 x 128) @ (128 x 2) + bias -> d_out (32 x 2)
__global__ void k_cls(const float* __restrict__ enc, const float* __restrict__ wcls,
                      const float* __restrict__ bcls, float* __restrict__ out) {
    int t = threadIdx.x;
    if (t < 64) {
        int b = t >> 1, cc = t & 1;
        float acc = bcls[cc];
#pragma unroll 4
        for (int z = 0; z < 128; ++z)
            acc += enc[b * 128 + z] * wcls[z * 2 + cc];
        out[t] = acc;
    }
}

// ---------------------------------------------------------------------------
static inline int nblk(size_t total, int threads) {
    size_t b = (total + (size_t)threads - 1) / threads;
    if (b > 32768) b = 32768;
    if (b < 1) b = 1;
    return (int)b;
}

template <int Fi, int Fo>
static void run_layer(const int* edge, const float* norm, const int* pidx,
                      const float* pval, const float* w, const float* bias,
                      int N, int nE, int nOut,
                      float* bufA, float* bufB, float* bufC, float* bufO,
                      hipStream_t stream) {
    constexpr int TILE_IN = 32 * Fi;
    constexpr int TILE_OUT = 32 * Fo;
    const int THR = 256;
    size_t tsz = (size_t)N * TILE_IN;
    size_t osz = (size_t)N * TILE_OUT;
    size_t wkstep = (size_t)Fi * Fo;
    int tasks = N * 2 * (Fo / 16);
    int ggrid = nblk((size_t)tasks * 32, THR);
    int sgrid = nblk((size_t)nE * (TILE_IN / 4), THR);

    float* t0 = bufA;
    float* t1 = bufB;
    float* t2 = bufC;

    // out = 0; out += T0 @ w[0]
    k_fill4<<<nblk(osz / 4, THR), THR, 0, stream>>>((float4*)bufO, osz / 4, 0.0f);
    k_cheb_gemm<Fi, Fo><<<ggrid, THR, 0, stream>>>(t0, w, bufO, N);

    // T1 = L @ T0 ; out += T1 @ w[1]
    k_fill4<<<nblk(tsz / 4, THR), THR, 0, stream>>>((float4*)t1, tsz / 4, 0.0f);
    k_scatter<TILE_IN><<<sgrid, THR, 0, stream>>>(edge, norm, t0, t1, nE, 1.0f);
    k_cheb_gemm<Fi, Fo><<<ggrid, THR, 0, stream>>>(t1, w + wkstep, bufO, N);

    // T_k = 2 L T_{k-1} - T_{k-2}; out += T_k @ w[k]
    float *tm2 = t0, *tm1 = t1, *cur = t2;
    for (int k = 2; k < 6; ++k) {
        k_scale_copy4<<<nblk(tsz / 4, THR), THR, 0, stream>>>(
            (float4*)cur, (const float4*)tm2, tsz / 4, -1.0f);
        k_scatter<TILE_IN><<<sgrid, THR, 0, stream>>>(edge, norm, tm1, cur, nE, 2.0f);
        k_cheb_gemm<Fi, Fo><<<ggrid, THR, 0, stream>>>(cur, w + k * wkstep, bufO, N);
        float* old = tm2; tm2 = tm1; tm1 = cur; cur = old;
    }

    k_bias_relu4<Fo><<<nblk(osz / 4, THR), THR, 0, stream>>>(bufO, bias, osz / 4);

    // Pool into bufA (next layer's node-major input)
    k_pool<TILE_OUT><<<nblk((size_t)nOut * (TILE_OUT / 4), THR), THR, 0, stream>>>(
        pidx, pval, bufO, bufA, nOut);
}

extern "C" void kernel_launch(void* const* d_in, const int* in_sizes, int n_in,
                              void* d_out, int out_size, void* d_ws, size_t ws_size,
                              hipStream_t stream) {
    (void)in_sizes; (void)n_in; (void)out_size; (void)ws_size;

    const float* x       = (const float*)d_in[0];
    const int*   edge[3] = {(const int*)d_in[1], (const int*)d_in[5], (const int*)d_in[9]};
    const float* norm[3] = {(const float*)d_in[2], (const float*)d_in[6], (const float*)d_in[10]};
    const int*   pidx[3] = {(const int*)d_in[3], (const int*)d_in[7], (const int*)d_in[11]};
    const float* pval[3] = {(const float*)d_in[4], (const float*)d_in[8], (const float*)d_in[12]};
    const float* w[3]    = {(const float*)d_in[13], (const float*)d_in[15], (const float*)d_in[17]};
    const float* bias[3] = {(const float*)d_in[14], (const float*)d_in[16], (const float*)d_in[18]};
    const float* wenc = (const float*)d_in[19];
    const float* benc = (const float*)d_in[20];
    const float* wcls = (const float*)d_in[21];
    const float* bcls = (const float*)d_in[22];

    // Scratch layout (floats): 3 rotation buffers (max N*32*Fin = 12500*512 = 6.4M),
    // output accumulator (max N*32*Fout = 50000*512 = 25.6M), encoder acc (32*128).
    const size_t ROT = 6400000;
    float* ws   = (float*)d_ws;
    float* bufA = ws;
    float* bufB = ws + ROT;
    float* bufC = ws + 2 * ROT;
    float* bufO = ws + 3 * ROT;
    float* bufE = ws + 3 * ROT + 25600000;

    const int THR = 256;

    // Layer-0 input -> node-major in bufA
    k_transpose_x<<<nblk((size_t)50000 * 96, THR), THR, 0, stream>>>(x, bufA, 50000);

    run_layer<3, 16>(edge[0], norm[0], pidx[0], pval[0], w[0], bias[0],
                     50000, 300000, 12500, bufA, bufB, bufC, bufO, stream);
    run_layer<16, 16>(edge[1], norm[1], pidx[1], pval[1], w[1], bias[1],
                      12500, 75000, 3125, bufA, bufB, bufC, bufO, stream);
    run_layer<16, 32>(edge[2], norm[2], pidx[2], pval[2], w[2], bias[2],
                      3125, 18750, 782, bufA, bufB, bufC, bufO, stream);

    // Split-K encoder GEMM (WMMA, 128 blocks x 16 waves) + bias/ReLU + classifier
    k_fill4<<<1, 256, 0, stream>>>((float4*)bufE, 1024, 0.0f);
    k_enc_partial<<<128, 512, 0, stream>>>(bufA, wenc, bufE);
    k_bias_relu4<128><<<1, 256, 0, stream>>>(bufE, benc, 1024);
    k_cls<<<1, 64, 0, stream>>>(bufE, wcls, bcls, (float*)d_out);
}